// FastFeedForward_987842478924
// MI455X (gfx1250) — compile-verified
//
#include <hip/hip_runtime.h>
#include <hip/hip_bf16.h>

// ---------------- problem constants ----------------
#define DEPTH     4
#define N_LEAVES  16
#define N_NODES   15
#define IN_F      512
#define LEAF_F    256
#define OUT_F     512
#define BATCH     2048

// 32-row tiles: two 16-row WMMA fragments per wave share each B fragment,
// halving W traffic from L2 versus 16-row tiles.
#define TILE_M    32
#define MAX_TILES (BATCH / TILE_M + (N_LEAVES - 1))   // 64 + 15 = 79
#define PAD_MAX   (MAX_TILES * TILE_M)                // 2528

// padded LDS strides (stride mod 64 banks == 4 -> conflict-free column walks)
#define XS (IN_F + 4)     // 516
#define HS (LEAF_F + 4)   // 260

typedef float v2f __attribute__((ext_vector_type(2)));
typedef float v8f __attribute__((ext_vector_type(8)));

// =====================================================================
// Kernel 1: tree routing. One wave32 per sample.
// =====================================================================
__global__ __launch_bounds__(256) void fff_route(
    const float* __restrict__ x, const float* __restrict__ nw,
    const float* __restrict__ nb, int* __restrict__ leaf) {
  const int wave = (blockIdx.x * blockDim.x + threadIdx.x) >> 5;
  const int lane = threadIdx.x & 31;
  if (wave >= BATCH) return;

  const float* xr = x + (size_t)wave * IN_F;
  float xs[IN_F / 32];
#pragma unroll
  for (int j = 0; j < IN_F / 32; ++j) xs[j] = xr[lane + 32 * j];

  int cur = 0;
#pragma unroll
  for (int d = 0; d < DEPTH; ++d) {
    const float* w = nw + (size_t)cur * IN_F;
    float s = 0.f;
#pragma unroll
    for (int j = 0; j < IN_F / 32; ++j) s += xs[j] * w[lane + 32 * j];
#pragma unroll
    for (int off = 16; off > 0; off >>= 1) s += __shfl_xor(s, off, 32);
    const int choice = (s + nb[cur]) > 0.f ? 1 : 0;
    cur = 2 * cur + choice + 1;
  }
  if (lane == 0) leaf[wave] = cur - N_NODES;
}

// =====================================================================
// Kernel 2: bucket samples by leaf, pad buckets to multiples of TILE_M.
// Single block of 256 threads.
// =====================================================================
__global__ __launch_bounds__(256) void fff_bucket(
    const int* __restrict__ leaf, int* __restrict__ sorted_idx,
    int* __restrict__ tile_leaf) {
  __shared__ int s_count[N_LEAVES];
  __shared__ int s_cursor[N_LEAVES];
  const int tid = threadIdx.x;

  if (tid < N_LEAVES) s_count[tid] = 0;
  __syncthreads();

  for (int b = tid; b < BATCH; b += 256) atomicAdd(&s_count[leaf[b]], 1);
  for (int i = tid; i < PAD_MAX; i += 256) sorted_idx[i] = -1;
  __syncthreads();

  if (tid == 0) {
    int off = 0;
    for (int l = 0; l < N_LEAVES; ++l) {
      s_cursor[l] = off;
      const int pc = (s_count[l] + TILE_M - 1) & ~(TILE_M - 1);
      for (int t = off / TILE_M; t < (off + pc) / TILE_M; ++t) tile_leaf[t] = l;
      off += pc;
    }
    for (int t = off / TILE_M; t < MAX_TILES; ++t) tile_leaf[t] = -1;
  }
  __syncthreads();

  for (int b = tid; b < BATCH; b += 256) {
    const int l = leaf[b];
    const int pos = atomicAdd(&s_cursor[l], 1);
    sorted_idx[pos] = b;
  }
}

// =====================================================================
// Kernel 3: fused 2-layer leaf FFN on 32-row tiles via fp32 WMMA.
// One block (8 waves, 256 threads) per row tile; each wave keeps two
// stacked 16x16 C fragments so every B fragment feeds 2 WMMAs.
//   layer1: [32 x 512] @ [512 x 256] + b1 -> hidden in LDS
//   layer2: [32 x 256] @ [256 x 512] + b2 -> scatter to out
// =====================================================================
__global__ __launch_bounds__(256) void fff_leaf_ffn(
    const float* __restrict__ x,
    const float* __restrict__ lw1, const float* __restrict__ lb1,
    const float* __restrict__ lw2, const float* __restrict__ lb2,
    const int* __restrict__ sorted_idx, const int* __restrict__ tile_leaf,
    float* __restrict__ out) {
  __shared__ __align__(16) float s_x[TILE_M * XS];   // 66,048 B
  __shared__ __align__(16) float s_h[TILE_M * HS];   // 33,280 B
  __shared__ int s_ridx[TILE_M];

  const int t  = blockIdx.x;
  const int lf = tile_leaf[t];
  if (lf < 0) return;                                 // uniform exit

  const int tid  = threadIdx.x;
  const int wv   = tid >> 5;
  const int lane = tid & 31;
  const int lo   = lane & 15;   // N column / A row within fragment
  const int hi   = lane >> 4;   // K-pair / row-half selector

  if (tid < TILE_M) s_ridx[tid] = sorted_idx[t * TILE_M + tid];
  __syncthreads();

  // gather the 32 rows of x into LDS (pad rows -> zeros)
  for (int i = tid; i < TILE_M * IN_F; i += 256) {
    const int r = i >> 9;          // i / 512
    const int c = i & (IN_F - 1);  // i % 512
    const int rb = s_ridx[r];
    s_x[r * XS + c] = (rb >= 0) ? x[(size_t)rb * IN_F + c] : 0.f;
  }
  __syncthreads();

  // ---------------- layer 1 ----------------
  const float* W1 = lw1 + (size_t)lf * IN_F * LEAF_F;
  const float* B1 = lb1 + (size_t)lf * LEAF_F;
  for (int nt = wv; nt < LEAF_F / 16; nt += 8) {
    const int n = nt * 16 + lo;
    v8f acc0, acc1;
    const float bias = B1[n];
#pragma unroll
    for (int e = 0; e < 8; ++e) { acc0[e] = bias; acc1[e] = bias; }
#pragma unroll 4
    for (int k0 = 0; k0 < IN_F; k0 += 4) {
      const int ka = k0 + 2 * hi;
      v2f a0 = *(const v2f*)(&s_x[lo * XS + ka]);         // rows 0..15
      v2f a1 = *(const v2f*)(&s_x[(lo + 16) * XS + ka]);  // rows 16..31
      v2f b;
      b.x = W1[(size_t)ka * LEAF_F + n];                  // B[k=ka  ][n]
      b.y = W1[(size_t)(ka + 1) * LEAF_F + n];            // B[k=ka+1][n]
      acc0 = __builtin_amdgcn_wmma_f32_16x16x4_f32(
          false, a0, false, b, (short)0, acc0, false, false);
      acc1 = __builtin_amdgcn_wmma_f32_16x16x4_f32(
          false, a1, false, b, (short)0, acc1, false, false);
    }
#pragma unroll
    for (int v = 0; v < 8; ++v) {
      s_h[(v + 8 * hi) * HS + n]        = acc0[v];
      s_h[(16 + v + 8 * hi) * HS + n]   = acc1[v];
    }
  }
  __syncthreads();

  // ---------------- layer 2 ----------------
  const float* W2 = lw2 + (size_t)lf * LEAF_F * OUT_F;
  const float* B2 = lb2 + (size_t)lf * OUT_F;
  for (int nt = wv; nt < OUT_F / 16; nt += 8) {
    const int n = nt * 16 + lo;
    v8f acc0, acc1;
    const float bias = B2[n];
#pragma unroll
    for (int e = 0; e < 8; ++e) { acc0[e] = bias; acc1[e] = bias; }
#pragma unroll 4
    for (int k0 = 0; k0 < LEAF_F; k0 += 4) {
      const int ka = k0 + 2 * hi;
      v2f a0 = *(const v2f*)(&s_h[lo * HS + ka]);
      v2f a1 = *(const v2f*)(&s_h[(lo + 16) * HS + ka]);
      v2f b;
      b.x = W2[(size_t)ka * OUT_F + n];
      b.y = W2[(size_t)(ka + 1) * OUT_F + n];
      acc0 = __builtin_amdgcn_wmma_f32_16x16x4_f32(
          false, a0, false, b, (short)0, acc0, false, false);
      acc1 = __builtin_amdgcn_wmma_f32_16x16x4_f32(
          false, a1, false, b, (short)0, acc1, false, false);
    }
#pragma unroll
    for (int v = 0; v < 8; ++v) {
      const int rb0 = s_ridx[v + 8 * hi];
      const int rb1 = s_ridx[16 + v + 8 * hi];
      if (rb0 >= 0) out[(size_t)rb0 * OUT_F + n] = acc0[v];
      if (rb1 >= 0) out[(size_t)rb1 * OUT_F + n] = acc1[v];
    }
  }
}

// =====================================================================
// launch
// =====================================================================
extern "C" void kernel_launch(void* const* d_in, const int* in_sizes, int n_in,
                              void* d_out, int out_size, void* d_ws, size_t ws_size,
                              hipStream_t stream) {
  const float* x   = (const float*)d_in[0];
  const float* nw  = (const float*)d_in[1];
  const float* nb  = (const float*)d_in[2];
  const float* lw1 = (const float*)d_in[3];
  const float* lb1 = (const float*)d_in[4];
  const float* lw2 = (const float*)d_in[5];
  const float* lb2 = (const float*)d_in[6];
  float* out = (float*)d_out;

  char* ws = (char*)d_ws;
  int* leaf       = (int*)(ws);                                   // 2048 ints
  int* sorted_idx = (int*)(ws + sizeof(int) * BATCH);             // 2528 ints
  int* tile_leaf  = (int*)(ws + sizeof(int) * (BATCH + PAD_MAX)); // 79 ints

  fff_route<<<BATCH / 8, 256, 0, stream>>>(x, nw, nb, leaf);
  fff_bucket<<<1, 256, 0, stream>>>(leaf, sorted_idx, tile_leaf);
  fff_leaf_ffn<<<MAX_TILES, 256, 0, stream>>>(x, lw1, lb1, lw2, lb2,
                                              sorted_idx, tile_leaf, out);
}